// FFM_Layer_58454504899040
// MI455X (gfx1250) — compile-verified
//
#include <hip/hip_runtime.h>

typedef __attribute__((ext_vector_type(2))) float v2f;
typedef __attribute__((ext_vector_type(8))) float v8f;

#define N_DENSE 13
#define N_SPARSE 26
#define FEAT_PER_SPARSE 1000
#define FIELD_NUM 39            // 13 + 26
#define K_DIM 16
#define VROW (FIELD_NUM * K_DIM)  // 624 floats per feature row of v

__global__ __launch_bounds__(256) void ffm_fused_kernel(
    const float* __restrict__ dense,    // [B, 13]
    const int*   __restrict__ sparse,   // [B, 26]
    const float* __restrict__ w0,       // [1]
    const float* __restrict__ w,        // [26013]
    const float* __restrict__ v,        // [26013, 39, 16] row-major
    float*       __restrict__ out,      // [B]
    int B)
{
    __shared__ float sA[16 * 16];          // dense tile, zero-padded 16x16
    __shared__ int   sIdx[16 * N_SPARSE];  // global feature ids per (row, j)
    __shared__ float sField[16 * VROW];    // field_f staging: [row][f*16+k]

    const int t  = threadIdx.x;
    const int b0 = blockIdx.x * 16;

    // ---- cooperative loads ---------------------------------------------
    {
        int row = t >> 4, d = t & 15;
        sA[t] = (d < N_DENSE) ? dense[(b0 + row) * N_DENSE + d] : 0.0f;
    }
    for (int i = t; i < 16 * N_SPARSE; i += 256) {
        int row = i / N_SPARSE, j = i % N_SPARSE;
        sIdx[i] = sparse[(b0 + row) * N_SPARSE + j] + N_DENSE + FEAT_PER_SPARSE * j;
    }
    __syncthreads();

    // ---- stage 1: dense einsum via f32 WMMA (16x16x4, 4 k-steps) -------
    const int lane = t & 31;
    const int wid  = t >> 5;
    const int m    = lane & 15;   // M index for A; N index for B/C/D
    const int kh   = lane >> 4;   // half-wave selector (K pairs / M+8 rows)

    // A fragments: A[m][d], lanes 0-15 hold K=4kk+{0,1}, lanes 16-31 K=4kk+{2,3}
    // sA is zero-padded for d >= 13, so padded K-steps contribute exactly 0.
    v2f afrag[4];
#pragma unroll
    for (int kk = 0; kk < 4; ++kk) {
        int d0 = 4 * kk + 2 * kh;
        afrag[kk][0] = sA[m * 16 + d0];
        afrag[kk][1] = sA[m * 16 + d0 + 1];
    }

    for (int f = wid; f < FIELD_NUM; f += 8) {   // wave-uniform loop
        const float* vf = v + f * K_DIM + m;     // B[d][n] = v[d*VROW + f*16 + n]
        // Unconditional loads for d = 0..15: rows 13..15 of v are valid
        // memory and are multiplied by the zero-padded A columns.
        v2f bfrag[4];
#pragma unroll
        for (int kk = 0; kk < 4; ++kk) {
            int d0 = 4 * kk + 2 * kh;
            bfrag[kk][0] = vf[(size_t)d0       * VROW];
            bfrag[kk][1] = vf[(size_t)(d0 + 1) * VROW];
        }
        v8f c = {};
#pragma unroll
        for (int kk = 0; kk < 4; ++kk)
            c = __builtin_amdgcn_wmma_f32_16x16x4_f32(
                    false, afrag[kk], false, bfrag[kk],
                    (short)0, c, false, false);
        // D layout: VGPR r -> row (kh*8 + r), col n = lane&15
#pragma unroll
        for (int r = 0; r < 8; ++r)
            sField[(kh * 8 + r) * VROW + f * K_DIM + m] = c[r];
    }
    __syncthreads();

    // ---- stage 2: sparse gathers + FM reduction ------------------------
    const int bb = t >> 4;   // batch row within tile
    const int k  = t & 15;   // embedding column

    float acc[FIELD_NUM];
#pragma unroll
    for (int f = 0; f < FIELD_NUM; ++f)
        acc[f] = sField[bb * VROW + f * K_DIM + k];

    for (int j = 0; j < N_SPARSE; ++j) {
        const float* vrow = v + (size_t)sIdx[bb * N_SPARSE + j] * VROW + k;
#pragma unroll
        for (int f = 0; f < FIELD_NUM; ++f)
            acc[f] += vrow[f * K_DIM];
    }

    float s = 0.0f, sq = 0.0f;
#pragma unroll
    for (int f = 0; f < FIELD_NUM; ++f) {
        s  += acc[f];
        sq += acc[f] * acc[f];
    }

    // first-order partial: dense part on lanes k<13, sparse j=k and j=k+16
    float first = w[sIdx[bb * N_SPARSE + k]];
    if (k < N_DENSE)        first += sA[bb * 16 + k] * w[k];
    if (k < N_SPARSE - 16)  first += w[sIdx[bb * N_SPARSE + k + 16]];

    float p = first + 0.5f * (s * s - sq);
#pragma unroll
    for (int off = 1; off < 16; off <<= 1)
        p += __shfl_xor(p, off, 32);

    if (k == 0)
        out[b0 + bb] = p + w0[0];
}

extern "C" void kernel_launch(void* const* d_in, const int* in_sizes, int n_in,
                              void* d_out, int out_size, void* d_ws, size_t ws_size,
                              hipStream_t stream) {
    const float* dense  = (const float*)d_in[0];
    const int*   sparse = (const int*)  d_in[1];
    const float* w0     = (const float*)d_in[2];
    const float* w      = (const float*)d_in[3];
    const float* v      = (const float*)d_in[4];
    float*       out    = (float*)d_out;

    int B = in_sizes[0] / N_DENSE;            // 4096
    ffm_fused_kernel<<<B / 16, 256, 0, stream>>>(dense, sparse, w0, w, v, out, B);
}